// OrthoLoss_3633542332712
// MI455X (gfx1250) — compile-verified
//
#include <hip/hip_runtime.h>
#include <hip/hip_bf16.h>

typedef float v2f __attribute__((ext_vector_type(2)));
typedef float v8f __attribute__((ext_vector_type(8)));

#define WAVES_PER_WG 8
#define WG_PER_GROUP 256
#define BLOCK_SIZE   (WAVES_PER_WG * 32)

// ---------------------------------------------------------------------------
// Zero the 64-float per-row accumulator in workspace (ws is poisoned 0xAA).
// ---------------------------------------------------------------------------
__global__ void ortho_zero_ws(float* __restrict__ dots) {
    dots[threadIdx.x] = 0.0f;
}

// ---------------------------------------------------------------------------
// Main streaming kernel: per-row dot products via V_WMMA_F32_16X16X4_F32.
//
// Row-group g handles batch rows [16g, 16g+16). For each K-chunk of 4:
//   A[m][k] = x1[16g+m][d0+k]   (16x4 f32)
//   B[k][n] = x2[16g+n][d0+k]   (4x16 f32)
//   C[m][n] += A x B            -> diagonal C[m][m] = partial dot of row m
//
// ISA A layout (16x4 f32): lanes 0-15 hold {K0,K1}, lanes 16-31 hold {K2,K3},
// M = lane%16. So each lane does ONE aligned b64 load per matrix at
//   base + row*D + d0 + 2*(lane/16)
// and x2 uses the identical per-lane address formula (B layout symmetric).
//
// Data is streamed exactly once (no reuse anywhere, 256 MiB > 192 MB L2),
// so all loads are non-temporal.
// ---------------------------------------------------------------------------
__global__ __launch_bounds__(BLOCK_SIZE) void ortho_dot_wmma(
    const float* __restrict__ x1, const float* __restrict__ x2,
    float* __restrict__ dots, int D, int chunksPerWave) {

    const int group = blockIdx.y;               // which 16-row group
    const int wave  = threadIdx.x >> 5;
    const int lane  = threadIdx.x & 31;

    const long long row  = (long long)(group * 16 + (lane & 15));
    const int       koff = (lane >> 4) << 1;    // 0 or 2

    const float* p1 = x1 + row * (long long)D + koff;
    const float* p2 = x2 + row * (long long)D + koff;

    // Blocked K assignment: each wave owns a contiguous run of K-chunks
    // (uniform loop count -> EXEC stays all-ones around the WMMAs).
    const long long gwave = (long long)blockIdx.x * WAVES_PER_WG + wave;
    long long d0 = gwave * (long long)chunksPerWave * 4;

    v8f c = {0.f, 0.f, 0.f, 0.f, 0.f, 0.f, 0.f, 0.f};

#pragma unroll 8
    for (int i = 0; i < chunksPerWave; ++i) {
        v2f a = __builtin_nontemporal_load((const v2f*)(p1 + d0)); // b64 th:NT
        v2f b = __builtin_nontemporal_load((const v2f*)(p2 + d0)); // b64 th:NT
        c = __builtin_amdgcn_wmma_f32_16x16x4_f32(
                /*neg_a=*/false, a, /*neg_b=*/false, b,
                /*c_mod=*/(short)0, c, /*reuse_a=*/false, /*reuse_b=*/false);
        d0 += 4;
    }

    // ---- extract diagonal C[m][m] ----
    // C/D layout: VGPR g = {lanes 0-15: M=g, N=lane} {lanes 16-31: M=g+8, N=lane-16}
    //   m in [0,8):  c[m]   at lane m
    //   m in [8,16): c[m-8] at lane m+16  (i.e. lanes 24..31)
    __shared__ float lred[16];
    if (threadIdx.x < 16) lred[threadIdx.x] = 0.0f;
    __syncthreads();

    int r = -1;
    if (lane < 8)        r = lane;        // rows 0..7
    else if (lane >= 24) r = lane - 16;   // rows 8..15
    if (r >= 0) {
        const int idx = r & 7;            // which accumulator VGPR holds the diag
        float v = c[0];
#pragma unroll
        for (int k = 1; k < 8; ++k)
            if (idx == k) v = c[k];       // cndmask chain, no scratch
        atomicAdd(&lred[r], v);           // ds_add_f32 across the 8 waves
    }
    __syncthreads();

    if (threadIdx.x < 16)                 // one global f32 atomic per row per WG
        atomicAdd(&dots[group * 16 + threadIdx.x], lred[threadIdx.x]);
}

// ---------------------------------------------------------------------------
// Finalize: loss = (1/(B*D)) * sum_b dots[b]^2   (B=64 threads)
// ---------------------------------------------------------------------------
__global__ void ortho_finalize(const float* __restrict__ dots,
                               float* __restrict__ out, float invDB) {
    __shared__ float s[64];
    const int t = threadIdx.x;
    const float d = dots[t];
    s[t] = d * d;
    __syncthreads();
#pragma unroll
    for (int off = 32; off > 0; off >>= 1) {
        if (t < off) s[t] += s[t + off];
        __syncthreads();
    }
    if (t == 0) out[0] = s[0] * invDB;
}

extern "C" void kernel_launch(void* const* d_in, const int* in_sizes, int n_in,
                              void* d_out, int out_size, void* d_ws, size_t ws_size,
                              hipStream_t stream) {
    const float* x1   = (const float*)d_in[0];
    const float* x2   = (const float*)d_in[1];
    float*       out  = (float*)d_out;
    float*       dots = (float*)d_ws;     // 64 floats of scratch

    const long long total = (long long)in_sizes[0];  // 64 * 524288
    const int B = 64;
    const int D = (int)(total / B);                  // 524288
    const int kChunks       = D / 4;                              // 131072
    const int totalWaves    = WG_PER_GROUP * WAVES_PER_WG;        // 2048
    const int chunksPerWave = kChunks / totalWaves;               // 64

    ortho_zero_ws<<<1, 64, 0, stream>>>(dots);

    dim3 grid(WG_PER_GROUP, B / 16);                 // 256 x 4 WGs
    ortho_dot_wmma<<<grid, BLOCK_SIZE, 0, stream>>>(x1, x2, dots, D, chunksPerWave);

    const float invDB = (float)(1.0 / ((double)D * (double)B));
    ortho_finalize<<<1, 64, 0, stream>>>(dots, out, invDB);
}